// Net_82506321756835
// MI455X (gfx1250) — compile-verified
//
#include <hip/hip_runtime.h>
#include <hip/hip_bf16.h>

#define N_NODES    40000
#define N_EDGES    640000
#define HIDDEN     128
#define NUM_BASIS  6
#define LAYERS     4
#define NUM_GRAPHS 256
#define NUM_TASKS  128

typedef __attribute__((ext_vector_type(2))) float v2f;
typedef __attribute__((ext_vector_type(4))) float v4f;
typedef __attribute__((ext_vector_type(8))) float v8f;

#define AS_STR 132   // LDS stride (floats) for A tiles: 528B rows keep 16B alignment, 4-bank skew
#define WFB    66    // floats per swizzled B-fragment block (32 lanes x v2f, +2 pad for banks)
// sWf holds one 32-K-row panel in WMMA fragment order: block index = ks*8 + nt (ks=k2/4)
#define WF_FLOATS (64 * WFB)   // 8 ks * 8 nt blocks

__device__ __forceinline__ float gelu_exact(float v) {
    return 0.5f * v * (1.0f + erff(v * 0.70710678118654752440f));
}

// ---------------------------------------------------------------------------
// Atom encoder: h[n,c] = sum_f atom_emb[f, x[n,f], c]
// ---------------------------------------------------------------------------
__global__ __launch_bounds__(128)
void k_atom(const int* __restrict__ x, const float* __restrict__ aemb,
            float* __restrict__ h)
{
    __shared__ int sx[9];
    const int c = threadIdx.x;
    for (int n = blockIdx.x; n < N_NODES; n += gridDim.x) {
        if (c < 9) sx[c] = x[n * 9 + c];
        __syncthreads();
        float s = 0.0f;
        #pragma unroll
        for (int f = 0; f < 9; ++f)
            s += aemb[((size_t)f * 64 + sx[f]) * HIDDEN + c];
        h[(size_t)n * HIDDEN + c] = s;
        __syncthreads();
    }
}

// ---------------------------------------------------------------------------
// filter stage 1: t1[e,c] = bases[e,:6] @ fe_W1[:,c] + fe_b1[c]
// ---------------------------------------------------------------------------
__global__ __launch_bounds__(128)
void k_filter1(const float* __restrict__ bases, const float* __restrict__ W1,
               const float* __restrict__ b1, float* __restrict__ out)
{
    __shared__ float sb[64 * NUM_BASIS];
    const int c = threadIdx.x;
    float wcol[NUM_BASIS];
    #pragma unroll
    for (int k = 0; k < NUM_BASIS; ++k) wcol[k] = W1[k * HIDDEN + c];
    const float bias = b1[c];
    const int nChunks = N_EDGES / 64;
    for (int ch = blockIdx.x; ch < nChunks; ch += gridDim.x) {
        const int e0 = ch * 64;
        for (int i = c; i < 64 * NUM_BASIS; i += 128)
            sb[i] = bases[(size_t)e0 * NUM_BASIS + i];
        __syncthreads();
        for (int r = 0; r < 64; ++r) {
            float s = bias;
            #pragma unroll
            for (int k = 0; k < NUM_BASIS; ++k) s += wcol[k] * sb[r * NUM_BASIS + k];
            out[(size_t)(e0 + r) * HIDDEN + c] = s;
        }
        __syncthreads();
    }
}

// ---------------------------------------------------------------------------
// column sum / sumsq over rows (for BatchNorm batch stats)
// ---------------------------------------------------------------------------
__global__ __launch_bounds__(128)
void k_colstats(const float* __restrict__ X, int rows, float* __restrict__ stats)
{
    const int c = threadIdx.x;
    float s = 0.0f, q = 0.0f;
    for (int r = blockIdx.x; r < rows; r += gridDim.x) {
        float v = X[(size_t)r * HIDDEN + c];
        s += v; q += v * v;
    }
    atomicAdd(stats + c, s);
    atomicAdd(stats + HIDDEN + c, q);
}

// act: 0=none 1=gelu 2=relu ; optional residual add (res may be null)
__global__ __launch_bounds__(128)
void k_bn_act(const float* __restrict__ X, float* __restrict__ Y,
              const float* __restrict__ g, const float* __restrict__ be,
              const float* __restrict__ stats, int rows, int act,
              const float* __restrict__ res)
{
    const int c = threadIdx.x;
    const float inv_n = 1.0f / (float)rows;
    const float mean = stats[c] * inv_n;
    const float var  = stats[HIDDEN + c] * inv_n - mean * mean;
    const float sc = rsqrtf(var + 1e-5f) * g[c];
    const float sh = be[c] - mean * sc;
    for (int r = blockIdx.x; r < rows; r += gridDim.x) {
        const size_t idx = (size_t)r * HIDDEN + c;
        float v = X[idx] * sc + sh;
        if (act == 1) v = gelu_exact(v);
        else if (act == 2) v = fmaxf(v, 0.0f);
        if (res) v += res[idx];
        Y[idx] = v;
    }
}

// ---------------------------------------------------------------------------
// Stage a 32-K-row panel of W (rows p*32..p*32+31, 128 cols) into LDS in
// WMMA B-fragment order: lane L of fragment (ks,nt) holds
//   { W[4*ks + 2*(L>>4)][nt*16 + (L&15)], W[4*ks + 2*(L>>4) + 1][...] }
// One aligned ds_load_b64 per fragment at compute time, bank-conflict free.
// ---------------------------------------------------------------------------
__device__ __forceinline__
void stage_w_panel(float* __restrict__ sWf, const float* __restrict__ W,
                   int p, int tid)
{
    for (int q = tid; q < 32 * 32; q += 128) {
        const int kk = q >> 5;          // panel K row 0..31
        const int c4 = (q & 31) << 2;   // col 0,4,...,124
        v4f v = *(const v4f*)(W + (size_t)(p * 32 + kk) * HIDDEN + c4);
        const int ks = kk >> 2, kr = kk & 3;
        const int nt = c4 >> 4, m0 = c4 & 15;
        const int base = (ks * 8 + nt) * WFB + (16 * (kr >> 1) + m0) * 2 + (kr & 1);
        sWf[base + 0] = v.x;
        sWf[base + 2] = v.y;
        sWf[base + 4] = v.z;
        sWf[base + 6] = v.w;
    }
}

// ---------------------------------------------------------------------------
// generic 128-wide WMMA GEMM: Y[rows,128] = X[rows,128] @ W[128,128] + bias
// wave-per-16-row strip, fp32 WMMA 16x16x4, LDS-staged A tiles + W fragments
// ---------------------------------------------------------------------------
__global__ __launch_bounds__(128)
void k_gemm128(const float* __restrict__ X, const float* __restrict__ W,
               const float* __restrict__ bias, float* __restrict__ Y, int rows)
{
    __shared__ float sWf[WF_FLOATS];
    __shared__ float sA[4][16 * AS_STR];
    const int tid  = threadIdx.x;
    const int w    = tid >> 5;
    const int lane = tid & 31;
    const int m    = lane & 15;
    const int hi   = lane >> 4;
    const int nStrips = rows >> 4;
    const int nGroups = (nStrips + 3) >> 2;

    for (int grp = blockIdx.x; grp < nGroups; grp += gridDim.x) {
        const int strip = grp * 4 + w;
        const bool valid = strip < nStrips;
        const int row0 = strip << 4;

        if (valid) {
            for (int r = 0; r < 16; ++r) {
                v4f v = *(const v4f*)(X + (size_t)(row0 + r) * HIDDEN + lane * 4);
                *(v4f*)&sA[w][r * AS_STR + lane * 4] = v;
            }
        }
        v8f acc[8];
        #pragma unroll
        for (int t = 0; t < 8; ++t)
            #pragma unroll
            for (int i = 0; i < 8; ++i) acc[t][i] = 0.0f;

        __syncthreads();
        for (int p = 0; p < 4; ++p) {           // K panels of 32
            stage_w_panel(sWf, W, p, tid);
            __syncthreads();
            if (valid) {
                #pragma unroll
                for (int k2 = 0; k2 < 32; k2 += 4) {
                    v2f a = *(const v2f*)&sA[w][m * AS_STR + p * 32 + k2 + 2 * hi];
                    const float* bp = &sWf[(k2 >> 2) * 8 * WFB + lane * 2];
                    #pragma unroll
                    for (int nt = 0; nt < 8; ++nt) {
                        v2f b = *(const v2f*)(bp + nt * WFB);
                        acc[nt] = __builtin_amdgcn_wmma_f32_16x16x4_f32(
                            false, a, false, b, (short)0, acc[nt], false, false);
                    }
                }
            }
            __syncthreads();
        }
        if (valid) {
            #pragma unroll
            for (int nt = 0; nt < 8; ++nt) {
                const int n = nt * 16 + m;
                const float bv = bias ? bias[n] : 0.0f;
                #pragma unroll
                for (int i = 0; i < 8; ++i) {
                    const int r = row0 + i + 8 * hi;
                    Y[(size_t)r * HIDDEN + n] = acc[nt][i] + bv;
                }
            }
        }
    }
}

// ---------------------------------------------------------------------------
// message kernel (per layer), fully fused:
//   A-row = h[src[e]] + bond_embed(edge_attr[e])           (gather in staging)
//   T = A @ pre_W + pre_b
//   xee = gelu(T) * bw ; atomicAdd(aggr[dst[e]], xee)      (fused epilogue)
// ---------------------------------------------------------------------------
__global__ __launch_bounds__(128)
void k_msg(const float* __restrict__ h, const int* __restrict__ src,
           const int* __restrict__ dstv, const int* __restrict__ eattr,
           const float* __restrict__ bemb, const float* __restrict__ W,
           const float* __restrict__ bias, const float* __restrict__ bw,
           float* __restrict__ aggr)
{
    __shared__ float sWf[WF_FLOATS];
    __shared__ float sA[4][16 * AS_STR];
    const int tid  = threadIdx.x;
    const int w    = tid >> 5;
    const int lane = tid & 31;
    const int m    = lane & 15;
    const int hi   = lane >> 4;
    const int nStrips = N_EDGES / 16;          // 40000, multiple of 4
    const int nGroups = nStrips / 4;

    for (int grp = blockIdx.x; grp < nGroups; grp += gridDim.x) {
        const int strip = grp * 4 + w;
        const int e0 = strip * 16;

        // per-row metadata lives in lanes 0..15, broadcast via shfl
        const int se  = e0 + m;
        const int sIdx = src[se];
        const int a0 = eattr[(size_t)se * 3 + 0];
        const int a1 = eattr[(size_t)se * 3 + 1];
        const int a2 = eattr[(size_t)se * 3 + 2];

        for (int r = 0; r < 16; ++r) {
            const int ridx = __shfl(sIdx, r);
            const int r0 = __shfl(a0, r);
            const int r1 = __shfl(a1, r);
            const int r2 = __shfl(a2, r);
            v4f v  = *(const v4f*)(h    + (size_t)ridx * HIDDEN + lane * 4);
            v4f u0 = *(const v4f*)(bemb + (size_t)(r0)      * HIDDEN + lane * 4);
            v4f u1 = *(const v4f*)(bemb + (size_t)(16 + r1) * HIDDEN + lane * 4);
            v4f u2 = *(const v4f*)(bemb + (size_t)(32 + r2) * HIDDEN + lane * 4);
            v = v + u0 + u1 + u2;
            *(v4f*)&sA[w][r * AS_STR + lane * 4] = v;
        }

        v8f acc[8];
        #pragma unroll
        for (int t = 0; t < 8; ++t)
            #pragma unroll
            for (int i = 0; i < 8; ++i) acc[t][i] = 0.0f;

        __syncthreads();
        for (int p = 0; p < 4; ++p) {
            stage_w_panel(sWf, W, p, tid);
            __syncthreads();
            #pragma unroll
            for (int k2 = 0; k2 < 32; k2 += 4) {
                v2f a = *(const v2f*)&sA[w][m * AS_STR + p * 32 + k2 + 2 * hi];
                const float* bp = &sWf[(k2 >> 2) * 8 * WFB + lane * 2];
                #pragma unroll
                for (int nt = 0; nt < 8; ++nt) {
                    v2f b = *(const v2f*)(bp + nt * WFB);
                    acc[nt] = __builtin_amdgcn_wmma_f32_16x16x4_f32(
                        false, a, false, b, (short)0, acc[nt], false, false);
                }
            }
            __syncthreads();
        }

        int d8[8];
        #pragma unroll
        for (int i = 0; i < 8; ++i) d8[i] = dstv[e0 + i + 8 * hi];
        #pragma unroll
        for (int nt = 0; nt < 8; ++nt) {
            const int n = nt * 16 + m;
            const float bv = bias[n];
            #pragma unroll
            for (int i = 0; i < 8; ++i) {
                const int e = e0 + i + 8 * hi;
                float xv = gelu_exact(acc[nt][i] + bv);
                xv *= bw[(size_t)e * HIDDEN + n];
                atomicAdd(aggr + (size_t)d8[i] * HIDDEN + n, xv);
            }
        }
    }
}

// ---------------------------------------------------------------------------
// edge softmax (grouped by dst), ordered-uint float max encoding
// ---------------------------------------------------------------------------
__global__ void k_fill_u(unsigned* p, size_t n, unsigned v)
{
    for (size_t i = blockIdx.x * (size_t)blockDim.x + threadIdx.x; i < n;
         i += (size_t)gridDim.x * blockDim.x) p[i] = v;
}
__global__ void k_zero_f(float* p, size_t n)
{
    for (size_t i = blockIdx.x * (size_t)blockDim.x + threadIdx.x; i < n;
         i += (size_t)gridDim.x * blockDim.x) p[i] = 0.0f;
}

__global__ __launch_bounds__(128)
void k_smax(const float* __restrict__ bw, const int* __restrict__ dstv,
            unsigned* __restrict__ mbuf)
{
    const int c = threadIdx.x;
    for (int e = blockIdx.x; e < N_EDGES; e += gridDim.x) {
        const int d = dstv[e];
        unsigned u = __float_as_uint(bw[(size_t)e * HIDDEN + c]);
        u = (u >> 31) ? ~u : (u | 0x80000000u);
        atomicMax(mbuf + (size_t)d * HIDDEN + c, u);
    }
}

__global__ __launch_bounds__(128)
void k_sexp(float* __restrict__ bw, const int* __restrict__ dstv,
            const unsigned* __restrict__ mbuf, float* __restrict__ sbuf)
{
    const int c = threadIdx.x;
    for (int e = blockIdx.x; e < N_EDGES; e += gridDim.x) {
        const int d = dstv[e];
        const unsigned u = mbuf[(size_t)d * HIDDEN + c];
        const float mm = (u & 0x80000000u) ? __uint_as_float(u ^ 0x80000000u)
                                           : __uint_as_float(~u);
        const size_t idx = (size_t)e * HIDDEN + c;
        const float v = expf(bw[idx] - mm);
        bw[idx] = v;
        atomicAdd(sbuf + (size_t)d * HIDDEN + c, v);
    }
}

__global__ __launch_bounds__(128)
void k_sdiv(float* __restrict__ bw, const int* __restrict__ dstv,
            const float* __restrict__ sbuf)
{
    const int c = threadIdx.x;
    for (int e = blockIdx.x; e < N_EDGES; e += gridDim.x) {
        const int d = dstv[e];
        const size_t idx = (size_t)e * HIDDEN + c;
        bw[idx] = bw[idx] / (sbuf[(size_t)d * HIDDEN + c] + 1e-16f);
    }
}

__global__ void k_copy(float* __restrict__ d, const float* __restrict__ s, size_t n)
{
    for (size_t i = blockIdx.x * (size_t)blockDim.x + threadIdx.x; i < n;
         i += (size_t)gridDim.x * blockDim.x) d[i] = s[i];
}

__global__ __launch_bounds__(128)
void k_pool(const float* __restrict__ h, const int* __restrict__ gids,
            float* __restrict__ hg)
{
    const int c = threadIdx.x;
    for (int n = blockIdx.x; n < N_NODES; n += gridDim.x) {
        const int g = gids[n];
        atomicAdd(hg + (size_t)g * HIDDEN + c, h[(size_t)n * HIDDEN + c]);
    }
}

// ---------------------------------------------------------------------------
extern "C" void kernel_launch(void* const* d_in, const int* in_sizes, int n_in,
                              void* d_out, int out_size, void* d_ws, size_t ws_size,
                              hipStream_t stream)
{
    const int*   x         = (const int*)  d_in[0];
    const int*   edge_attr = (const int*)  d_in[1];
    const float* bases     = (const float*)d_in[2];
    const int*   src       = (const int*)  d_in[3];
    const int*   dst       = (const int*)  d_in[4];
    const int*   gids      = (const int*)  d_in[5];
    const float* atom_emb  = (const float*)d_in[6];
    const float* bond_emb  = (const float*)d_in[7];
    const float* fe_W1 = (const float*)d_in[8];
    const float* fe_b1 = (const float*)d_in[9];
    const float* fe_g1 = (const float*)d_in[10];
    const float* fe_be1= (const float*)d_in[11];
    const float* fe_W2 = (const float*)d_in[12];
    const float* fe_b2 = (const float*)d_in[13];
    const float* fe_g2 = (const float*)d_in[14];
    const float* fe_be2= (const float*)d_in[15];
    const float* pre_W = (const float*)d_in[16];
    const float* pre_b = (const float*)d_in[17];
    const float* ffn_W1= (const float*)d_in[18];
    const float* ffn_b1= (const float*)d_in[19];
    const float* ffn_g1= (const float*)d_in[20];
    const float* ffn_be1=(const float*)d_in[21];
    const float* ffn_W2= (const float*)d_in[22];
    const float* ffn_b2= (const float*)d_in[23];
    const float* ffn_g2= (const float*)d_in[24];
    const float* ffn_be2=(const float*)d_in[25];
    const float* head_W= (const float*)d_in[26];
    const float* head_b= (const float*)d_in[27];
    float* out = (float*)d_out;

    const size_t EH = (size_t)N_EDGES * HIDDEN;
    const size_t NH = (size_t)N_NODES * HIDDEN;
    float* ws   = (float*)d_ws;
    float* b1g  = ws;            // [E,H]
    float* bw   = b1g + EH;      // [E,H]
    float* h    = bw  + EH;      // [N,H]
    float* hx   = h   + NH;      // [N,H]  (also softmax max buffer pre-layers)
    float* y1   = hx  + NH;      // [N,H]  (also softmax sum buffer pre-layers)
    float* y2   = y1  + NH;      // [N,H]
    float* stats= y2  + NH;      // [256]
    float* hg   = stats + 256;   // [NUM_GRAPHS, H]

    const int EGRID = 4096, NGRID = 2048, SGRID = 512;
    const unsigned NEG_INF_ENC = 0x007FFFFFu;  // encode(-inf)

    // ---- encoders -------------------------------------------------------
    k_atom<<<2048, 128, 0, stream>>>(x, atom_emb, h);

    // ---- filter encoder -------------------------------------------------
    k_filter1<<<2048, 128, 0, stream>>>(bases, fe_W1, fe_b1, bw);     // t1 -> bw
    k_zero_f<<<64, 256, 0, stream>>>(stats, 256);
    k_colstats<<<SGRID, 128, 0, stream>>>(bw, N_EDGES, stats);
    k_bn_act<<<EGRID, 128, 0, stream>>>(bw, b1g, fe_g1, fe_be1, stats,
                                        N_EDGES, 1, nullptr);         // gelu -> b1g
    {
        int groups = (N_EDGES / 16 + 3) / 4;
        int grid = groups < 2048 ? groups : 2048;
        k_gemm128<<<grid, 128, 0, stream>>>(b1g, fe_W2, fe_b2, bw, N_EDGES);
    }
    k_zero_f<<<64, 256, 0, stream>>>(stats, 256);
    k_colstats<<<SGRID, 128, 0, stream>>>(bw, N_EDGES, stats);
    k_bn_act<<<EGRID, 128, 0, stream>>>(bw, bw, fe_g2, fe_be2, stats,
                                        N_EDGES, 1, nullptr);         // gelu in place

    // ---- edge softmax over dst -----------------------------------------
    k_fill_u<<<2048, 256, 0, stream>>>((unsigned*)hx, NH, NEG_INF_ENC);
    k_zero_f<<<2048, 256, 0, stream>>>(y1, NH);
    k_smax<<<EGRID, 128, 0, stream>>>(bw, dst, (unsigned*)hx);
    k_sexp<<<EGRID, 128, 0, stream>>>(bw, dst, (const unsigned*)hx, y1);
    k_sdiv<<<EGRID, 128, 0, stream>>>(bw, dst, y1);

    // ---- message-passing layers ----------------------------------------
    const int mgroups = (N_EDGES / 16) / 4;
    const int mgrid = mgroups < 2048 ? mgroups : 2048;
    const int ngroups = (N_NODES / 16 + 3) / 4;
    const int ngrid = ngroups < 2048 ? ngroups : 2048;

    for (int l = 0; l < LAYERS; ++l) {
        const float* Wl  = pre_W  + (size_t)l * HIDDEN * HIDDEN;
        const float* bl  = pre_b  + (size_t)l * HIDDEN;
        const float* W1l = ffn_W1 + (size_t)l * HIDDEN * HIDDEN;
        const float* b1l = ffn_b1 + (size_t)l * HIDDEN;
        const float* g1l = ffn_g1 + (size_t)l * HIDDEN;
        const float* be1l= ffn_be1+ (size_t)l * HIDDEN;
        const float* W2l = ffn_W2 + (size_t)l * HIDDEN * HIDDEN;
        const float* b2l = ffn_b2 + (size_t)l * HIDDEN;
        const float* g2l = ffn_g2 + (size_t)l * HIDDEN;
        const float* be2l= ffn_be2+ (size_t)l * HIDDEN;

        k_copy<<<2048, 256, 0, stream>>>(hx, h, NH);   // aggr starts at h
        k_msg<<<mgrid, 128, 0, stream>>>(h, src, dst, edge_attr, bond_emb,
                                         Wl, bl, bw, hx);   // hx = h + aggr

        k_gemm128<<<ngrid, 128, 0, stream>>>(hx, W1l, b1l, y1, N_NODES);
        k_zero_f<<<64, 256, 0, stream>>>(stats, 256);
        k_colstats<<<SGRID, 128, 0, stream>>>(y1, N_NODES, stats);
        k_bn_act<<<NGRID, 128, 0, stream>>>(y1, y1, g1l, be1l, stats,
                                            N_NODES, 2, nullptr);      // relu

        k_gemm128<<<ngrid, 128, 0, stream>>>(y1, W2l, b2l, y2, N_NODES);
        k_zero_f<<<64, 256, 0, stream>>>(stats, 256);
        k_colstats<<<SGRID, 128, 0, stream>>>(y2, N_NODES, stats);
        k_bn_act<<<NGRID, 128, 0, stream>>>(y2, h, g2l, be2l, stats,
                                            N_NODES, 2, hx);           // h = hx + relu(bn)
    }

    // ---- pooling + head -------------------------------------------------
    k_zero_f<<<64, 256, 0, stream>>>(hg, (size_t)NUM_GRAPHS * HIDDEN);
    k_pool<<<2048, 128, 0, stream>>>(h, gids, hg);
    {
        int groups = (NUM_GRAPHS / 16 + 3) / 4;
        k_gemm128<<<groups, 128, 0, stream>>>(hg, head_W, head_b, out, NUM_GRAPHS);
    }
}